// Contextual_Histogram_Binning_33818572488972
// MI455X (gfx1250) — compile-verified
//
#include <hip/hip_runtime.h>
#include <cstdint>

// ---------------------------------------------------------------------------
// Contextual histogram binning / calibration, fused for MI455X (gfx1250).
//
// Roofline: ~135 MB mandatory reads vs ~0.7 GFLOP -> bandwidth bound at
// 23.3 TB/s (~6us). Strategy: single pass over images/labels, LDS halo
// tiling (320KB/WGP), async global->LDS staging (ASYNCcnt path), LDS
// ds_add_f32 histogram with one global-atomic flush per block, and a
// WMMA f32 16x16x4 reduction for the label N-sum (bandwidth-neutral,
// exercises the matrix pipe).
// ---------------------------------------------------------------------------

#define HH 512
#define WW 512
#define HWPIX (512 * 512)
#define NSUP 32
#define NCH 3
#define NUM_BINS 15

typedef __attribute__((ext_vector_type(2))) float v2f;
typedef __attribute__((ext_vector_type(8))) float v8f;

__device__ __forceinline__ uint32_t lds_off_u32(const void* p) {
  // generic (flat) LDS address: low 32 bits == LDS byte offset (aperture rule)
  return (uint32_t)(uintptr_t)p;
}

__device__ __forceinline__ void async_g2l_b32(uint32_t lds_byte_off, const float* gaddr) {
  // CDNA5 async copy: each lane copies 4 bytes from the global address in vaddr to the LDS address in vdst, tracked by ASYNCcnt
  asm volatile("global_load_async_to_lds_b32 %0, %1, off"
               :: "v"(lds_byte_off), "v"(gaddr)
               : "memory");
}

__device__ __forceinline__ void wait_async0() {
  asm volatile("s_wait_asynccnt 0" ::: "memory");
}

// ---------------------------------------------------------------------------
// Kernel Z: zero the 32-float accumulator region (counts + positives).
// ---------------------------------------------------------------------------
__global__ void zero_accum_kernel(float* p, int n) {
  int i = blockIdx.x * blockDim.x + threadIdx.x;
  if (i < n) p[i] = 0.0f;
}

// ---------------------------------------------------------------------------
// Kernel A: sum_labs[p] = sum_n labs[n][p], via chained V_WMMA_F32_16X16X4_F32
// with A = ones(16x4).  C[m,n] = sum_k B[k,n]; column n == lane%16 (documented
// C/D layout), rows fed 4 planes per WMMA, 8 chained WMMAs cover N=32 planes.
// One wave produces 16 pixel sums; loads stay 64B-coalesced.
// ---------------------------------------------------------------------------
__global__ void sum_labels_kernel(const float* __restrict__ labs,
                                  float* __restrict__ sum_out) {
#if __has_builtin(__builtin_amdgcn_wmma_f32_16x16x4_f32)
  int gtid = blockIdx.x * blockDim.x + threadIdx.x;
  int wave = gtid >> 5;          // wave32
  int lane = threadIdx.x & 31;
  int col  = lane & 15;
  int half = lane >> 4;
  int p0   = wave * 16;          // 16 pixels per wave

  v2f a; a.x = 1.0f; a.y = 1.0f; // all-ones A => C rows are B column sums
  v8f c = {};
#pragma unroll
  for (int i = 0; i < 8; ++i) {
    int k0 = 4 * i + 2 * half;   // lanes 0-15 feed planes {4i,4i+1}, 16-31 {4i+2,4i+3}
    v2f b;
    b.x = labs[(size_t)k0 * HWPIX + p0 + col];
    b.y = labs[(size_t)(k0 + 1) * HWPIX + p0 + col];
    c = __builtin_amdgcn_wmma_f32_16x16x4_f32(false, a, false, b,
                                              (short)0, c, false, false);
  }
  if (half == 0) sum_out[p0 + col] = c[0];   // C[0, col] = sum over all 32 planes
#else
  int p = blockIdx.x * blockDim.x + threadIdx.x;
  if (p < HWPIX) {
    float s = 0.0f;
    for (int n = 0; n < NSUP; ++n) s += labs[(size_t)n * HWPIX + p];
    sum_out[p] = s;
  }
#endif
}

// ---------------------------------------------------------------------------
// Kernel C: fused per-plane tile kernel.
//  - 32x32 output tile, 36x36 input halo (conv halo 1 + pool halo 1)
//  - async global->LDS staging of {img c0..c2, labs[n], sum_labs} tiles
//  - conv3x3(img,w_img) + (conv3x3(sum,w_lab)-conv3x3(lab,w_lab))/31
//  - sigmoid, 3x3 valid-count neighbor mean, 15-bin LDS histogram
// ---------------------------------------------------------------------------
#define TS 32
#define IN 36      // TS + 4
#define PN 34      // TS + 2

__global__ void __launch_bounds__(256)
fused_support_kernel(const float* __restrict__ imgs,   // [N,3,H,W]
                     const float* __restrict__ labs,   // [N,1,H,W]
                     const float* __restrict__ ssum_g, // [H,W]
                     const float* __restrict__ wimg,   // [1,3,3,3]
                     const float* __restrict__ wlab,   // [1,1,3,3]
                     float* __restrict__ gcnt,         // [15]
                     float* __restrict__ gpos) {       // [15]
  __shared__ float simg[NCH][IN * IN];
  __shared__ float slab[IN * IN];
  __shared__ float ssum[IN * IN];
  __shared__ float sp[PN * PN];
  __shared__ float hc[NUM_BINS];
  __shared__ float hp[NUM_BINS];
  __shared__ float swt[36];   // [0..26]=w_img, [27..35]=w_lab

  const int tid = threadIdx.x;
  const int gx0 = blockIdx.x * TS;
  const int gy0 = blockIdx.y * TS;
  const int n   = blockIdx.z;

  const float* img_n = imgs + (size_t)n * NCH * HWPIX;
  const float* lab_n = labs + (size_t)n * HWPIX;

  if (tid < NUM_BINS) { hc[tid] = 0.0f; hp[tid] = 0.0f; }
  if (tid < 27) swt[tid] = wimg[tid];
  else if (tid < 36) swt[tid] = wlab[tid - 27];

  // ---- stage 5 halo tiles (3 img planes + lab + sum) via async LDS copies --
  for (int t = tid; t < IN * IN * 5; t += 256) {
    int src = t / (IN * IN);
    int rem = t - src * (IN * IN);
    int r = rem / IN;
    int cc = rem - r * IN;
    int gy = gy0 - 2 + r;
    int gx = gx0 - 2 + cc;
    bool inb = (gy >= 0) && (gy < HH) && (gx >= 0) && (gx < WW);
    float* dst;
    const float* gsrc;
    if (src < NCH) {
      dst = &simg[src][rem];
      gsrc = img_n + (size_t)src * HWPIX + gy * WW + gx;
    } else if (src == NCH) {
      dst = &slab[rem];
      gsrc = lab_n + gy * WW + gx;
    } else {
      dst = &ssum[rem];
      gsrc = ssum_g + gy * WW + gx;
    }
    if (inb) async_g2l_b32(lds_off_u32(dst), gsrc);   // zero-pad outside image
    else     *dst = 0.0f;
  }
  wait_async0();
  __syncthreads();

  // ---- conv + sigmoid over the 34x34 (pool-halo) region ---------------------
  const float inv_nm1 = 1.0f / (float)(NSUP - 1);
  for (int t = tid; t < PN * PN; t += 256) {
    int i = t / PN;           // tile-rel row of p-region (-1 offset vs output)
    int j = t - i * PN;
    int gy = gy0 - 1 + i;
    int gx = gx0 - 1 + j;
    float acc = 0.0f;
#pragma unroll
    for (int ch = 0; ch < NCH; ++ch)
#pragma unroll
      for (int dy = 0; dy < 3; ++dy)
#pragma unroll
        for (int dx = 0; dx < 3; ++dx)
          acc = fmaf(swt[ch * 9 + dy * 3 + dx],
                     simg[ch][(i + dy) * IN + (j + dx)], acc);
    float accl = 0.0f;
#pragma unroll
    for (int dy = 0; dy < 3; ++dy)
#pragma unroll
      for (int dx = 0; dx < 3; ++dx) {
        int idx = (i + dy) * IN + (j + dx);
        accl = fmaf(swt[27 + dy * 3 + dx], ssum[idx] - slab[idx], accl);
      }
    acc += accl * inv_nm1;                     // conv of LOO mean by linearity
    float pr = 1.0f / (1.0f + __expf(-acc));   // sigmoid
    bool vld = (gy >= 0) && (gy < HH) && (gx >= 0) && (gx < WW);
    sp[t] = vld ? pr : 0.0f;
  }
  __syncthreads();

  // ---- neighbor mean (count_include_pad=False) + histogram ------------------
  for (int t = tid; t < TS * TS; t += 256) {
    int oy = t / TS;
    int ox = t - oy * TS;
    int gy = gy0 + oy;
    int gx = gx0 + ox;
    float s = 0.0f, cnt = 0.0f;
#pragma unroll
    for (int dy = -1; dy <= 1; ++dy)
#pragma unroll
      for (int dx = -1; dx <= 1; ++dx) {
        int ny = gy + dy, nx = gx + dx;
        if (ny >= 0 && ny < HH && nx >= 0 && nx < WW) {
          s += sp[(oy + 1 + dy) * PN + (ox + 1 + dx)];
          cnt += 1.0f;
        }
      }
    float yp = s / cnt;
    int b = (int)(yp * (float)NUM_BINS);
    b = b < 0 ? 0 : (b > NUM_BINS - 1 ? NUM_BINS - 1 : b);
    float yt = (slab[(oy + 2) * IN + (ox + 2)] > 0.5f) ? 1.0f : 0.0f;
    atomicAdd(&hc[b], 1.0f);   // ds_add_f32
    atomicAdd(&hp[b], yt);
  }
  __syncthreads();

  if (tid < NUM_BINS) {
    atomicAdd(&gcnt[tid], hc[tid]);   // global_atomic_add_f32, 30 per block
    atomicAdd(&gpos[tid], hp[tid]);
  }
}

// ---------------------------------------------------------------------------
// Kernel D: target sigmoid -> bin -> gather bin frequency.
// ---------------------------------------------------------------------------
__global__ void finalize_kernel(const float* __restrict__ tlogits,
                                const float* __restrict__ gcnt,
                                const float* __restrict__ gpos,
                                float* __restrict__ out) {
  int p = blockIdx.x * blockDim.x + threadIdx.x;
  if (p < HWPIX) {
    float pr = 1.0f / (1.0f + __expf(-tlogits[p]));
    int b = (int)(pr * (float)NUM_BINS);
    b = b < 0 ? 0 : (b > NUM_BINS - 1 ? NUM_BINS - 1 : b);
    float c = gcnt[b];
    float po = gpos[b];
    out[p] = (c > 0.0f) ? (po / fmaxf(c, 1.0f)) : 0.0f;
  }
}

// ---------------------------------------------------------------------------
extern "C" void kernel_launch(void* const* d_in, const int* in_sizes, int n_in,
                              void* d_out, int out_size, void* d_ws, size_t ws_size,
                              hipStream_t stream) {
  (void)in_sizes; (void)n_in; (void)out_size; (void)ws_size;
  const float* imgs = (const float*)d_in[0];   // [1,32,3,512,512]
  const float* labs = (const float*)d_in[1];   // [1,32,1,512,512]
  const float* tlog = (const float*)d_in[2];   // [1,1,512,512]
  const float* wimg = (const float*)d_in[3];   // [1,3,3,3]
  const float* wlab = (const float*)d_in[4];   // [1,1,3,3]
  float* out = (float*)d_out;                  // [1,1,512,512]

  float* ssum = (float*)d_ws;                  // HWPIX floats
  float* gcnt = ssum + HWPIX;                  // 15 floats (padded to 16)
  float* gpos = gcnt + 16;                     // 15 floats

  zero_accum_kernel<<<1, 64, 0, stream>>>(gcnt, 32);

  // 2048 blocks * 8 waves * 16 px = 262144 px (exact)
  sum_labels_kernel<<<2048, 256, 0, stream>>>(labs, ssum);

  dim3 gridC(WW / TS, HH / TS, NSUP);          // (16,16,32)
  fused_support_kernel<<<gridC, 256, 0, stream>>>(imgs, labs, ssum, wimg, wlab,
                                                  gcnt, gpos);

  finalize_kernel<<<HWPIX / 256, 256, 0, stream>>>(tlog, gcnt, gpos, out);
}